// FourierCDE_27487790694625
// MI455X (gfx1250) — compile-verified
//
#include <hip/hip_runtime.h>
#include <stdint.h>

// ---------------------------------------------------------------------------
// FourierCDE on gfx1250 (MI455X).
// Pipeline:
//   k_init_dft : build fftshift-folded DFT matrices as WMMA A-fragments (ws)
//   k_pre      : Fu0 = A z0, G = Gw z0
//   k_wfrag    : pre-swizzle W[x] (Re,Im) into WMMA A-fragment layout (ws)
//   k_z0ft     : z_init = fftshift(fft(z0)) (direct DFT, tiny)
//   k_fwd      : Hu = Fu0 + sum_f G*xi, fused with DFT GEMM via WMMA f32;
//                F-fragments staged into LDS with TENSOR_LOAD_TO_LDS
//   k_scan     : per-x sequential CDE scan; W[x] fragments (256KB) DMA'd to
//                LDS via TDM; per step 2 GEMMs (32x1024)x(1024x16) with
//                v_wmma_f32_16x16x4_f32; z trajectory written in place
//   k_inv      : inverse DFT GEMM via WMMA, real part -> d_out
// ---------------------------------------------------------------------------

typedef float v2f __attribute__((ext_vector_type(2)));
typedef float v8f __attribute__((ext_vector_type(8)));

#define PI_F 3.14159265358979323846f

// workspace layout in floats
#define WS_FFWD   0          // 32768    : forward DFT A-fragments
#define WS_FINV   32768      // 32768    : inverse DFT A-fragments
#define WS_FU0    65536      // 32768    : Fu0[b,i,x]
#define WS_G      98304      // 262144   : G[b,i,f,x]
#define WS_ZINIT  360448     // 65536    : z_init (B,X,H) float2
#define WS_HFT    425984     // 16777216 : H_ft / z_all (B,X,T,H) float2
#define WS_WFRAG  17203200   // 8388608  : W A-fragments, 65536 per x

static __device__ inline v8f wmma_f32(v2f a, v2f b, v8f c) {
  return __builtin_amdgcn_wmma_f32_16x16x4_f32(false, a, false, b, (short)0, c,
                                               false, false);
}

// ---------------------------------------------------------------------------
// Tensor Data Mover: contiguous global -> LDS copy of rows*epr f32 values.
// D# per CDNA5 ISA.
#if defined(__has_builtin)
#if __has_builtin(__builtin_amdgcn_tensor_load_to_lds) && \
    __has_builtin(__builtin_amdgcn_s_wait_tensorcnt)
#define USE_TDM 1
#endif
#endif

#ifdef USE_TDM
typedef unsigned int u32x4 __attribute__((ext_vector_type(4)));
typedef int i32x4 __attribute__((ext_vector_type(4)));
typedef int i32x8 __attribute__((ext_vector_type(8)));

static __device__ inline void tdm_copy_f32(const float* gsrc, void* ldsDst,
                                           unsigned epr, unsigned rows) {
  unsigned long long ga = (unsigned long long)(uintptr_t)gsrc;
  unsigned ldsByte = (unsigned)(uintptr_t)ldsDst;  // low 32 bits = LDS offset
  u32x4 g0;
  g0[0] = 1u;                                   // count=1 (valid user D#)
  g0[1] = ldsByte;                              // lds_addr (bytes)
  g0[2] = (unsigned)(ga & 0xFFFFFFFFu);         // global_addr[31:0]
  g0[3] = (unsigned)((ga >> 32) & 0x1FFFFFFu)   // global_addr[56:32]
          | (2u << 30);                         // type=2 ("image")
  i32x8 g1;
  g1[0] = 0x20000;                              // data_size=2 (4B), no flags
  g1[1] = (int)((epr & 0xFFFFu) << 16);         // tensor_dim0[15:0]
  g1[2] = (int)(((epr >> 16) & 0xFFFFu) | ((rows & 0xFFFFu) << 16));
  g1[3] = (int)(((rows >> 16) & 0xFFFFu) | ((epr & 0xFFFFu) << 16)); // tile_dim0
  g1[4] = (int)(rows & 0xFFFFu);                // tile_dim1 (tile_dim2=0)
  g1[5] = (int)epr;                             // tensor_dim0_stride[31:0]
  g1[6] = 0;
  g1[7] = 0;
  i32x4 gz = {0, 0, 0, 0};
#if __clang_major__ >= 23
  i32x8 gz8 = {0, 0, 0, 0, 0, 0, 0, 0};
  __builtin_amdgcn_tensor_load_to_lds(g0, g1, gz, gz, gz8, 0);
#else
  __builtin_amdgcn_tensor_load_to_lds(g0, g1, gz, gz, 0);
#endif
}
#define TDM_WAIT() __builtin_amdgcn_s_wait_tensorcnt(0)
#endif

// ---------------------------------------------------------------------------
// DFT matrices as A-fragments.
// Fragment flat layout per array: ((g*8+m)*32+kt)*64 + lane*2 + p
//   A element: row = 16m + (lane&15), k = 4kt + 2*(lane>>4) + p
// Forward:  F[r,x]     = exp(-2*pi*i*((r+64)&127)*x/128)          (fftshift rows)
// Inverse:  Fi[x,kidx] = exp(+2*pi*i*x*((kidx+64)&127)/128)/128   (ifftshift cols)
__global__ void __launch_bounds__(256) k_init_dft(float* ws) {
  int idx = blockIdx.x * 256 + threadIdx.x;   // 0..65535
  int arr = idx >> 15;
  int rem = idx & 32767;
  int p    = rem & 1;
  int lane = (rem >> 1) & 31;
  int kt   = (rem >> 6) & 31;
  int m    = (rem >> 11) & 7;
  int g    = rem >> 14;
  int row = 16 * m + (lane & 15);
  int col = 4 * kt + 2 * (lane >> 4) + p;
  float val;
  if (arr == 0) {
    int kk = (row + 64) & 127;
    int ph = (kk * col) & 127;                 // exact phase reduction
    float th = (2.0f * PI_F / 128.0f) * (float)ph;
    val = (g == 0) ? __cosf(th) : -__sinf(th);
  } else {
    int kk = (col + 64) & 127;
    int ph = (row * kk) & 127;
    float th = (2.0f * PI_F / 128.0f) * (float)ph;
    val = ((g == 0) ? __cosf(th) : __sinf(th)) * (1.0f / 128.0f);
  }
  ws[idx] = val;
}

// ---------------------------------------------------------------------------
// Fu0[b,i,x] = sum_h A[i,h] z0[b,h,x];  G[b,i,f,x] = sum_h Gw[i,f,h] z0[b,h,x]
__global__ void __launch_bounds__(256) k_pre(const float* __restrict__ A,
                                             const float* __restrict__ Gw,
                                             const float* __restrict__ z0,
                                             float* ws) {
  int idx = blockIdx.x * 256 + threadIdx.x;
  float* Fu0 = ws + WS_FU0;
  float* G   = ws + WS_G;
  if (idx < 32768) {
    int x = idx & 127, i = (idx >> 7) & 31, b = idx >> 12;
    float s = 0.f;
    for (int h = 0; h < 32; ++h) s += A[i * 32 + h] * z0[(b * 32 + h) * 128 + x];
    Fu0[idx] = s;
  } else if (idx < 32768 + 262144) {
    int e = idx - 32768;
    int x = e & 127, f = (e >> 7) & 7, i = (e >> 10) & 31, b = e >> 15;
    float s = 0.f;
    for (int h = 0; h < 32; ++h)
      s += Gw[(i * 8 + f) * 32 + h] * z0[(b * 32 + h) * 128 + x];
    G[e] = s;
  }
}

// ---------------------------------------------------------------------------
// Pre-swizzle W into WMMA A-fragment layout (per x: 65536 floats).
// s = ((g*2+m)*256+kt)*64 + lane*2 + p ; element (i = 16m+(lane&15),
// k = 4kt + 2*(lane>>4) + p) of W2[i][k] = W[x, i, k/32, k%32].
__global__ void __launch_bounds__(256) k_wfrag(const float* __restrict__ Wr,
                                               const float* __restrict__ Wi,
                                               float* ws) {
  int idx = blockIdx.x * 256 + threadIdx.x;   // 0..8388607
  int x = idx >> 16;
  int s = idx & 65535;
  int p = s & 1;
  int lane = (s >> 1) & 31;
  int kt = (s >> 6) & 255;
  int m = (s >> 14) & 1;
  int g = s >> 15;
  int i = m * 16 + (lane & 15);
  int k = kt * 4 + 2 * (lane >> 4) + p;
  const float* Wg = g ? Wi : Wr;
  ws[WS_WFRAG + idx] = Wg[x * 32768 + i * 1024 + k];
}

// ---------------------------------------------------------------------------
// z_init[b,k,h] = fftshift(fft_x(z0[b,h,:]))[k]   (256 tiny DFTs)
__global__ void __launch_bounds__(128) k_z0ft(const float* __restrict__ z0,
                                              float* ws) {
  __shared__ float row[128];
  int b = blockIdx.x >> 5, h = blockIdx.x & 31;
  int k = threadIdx.x;
  row[k] = z0[(b * 32 + h) * 128 + k];
  __syncthreads();
  int kk = (k + 64) & 127;
  float sr = 0.f, si = 0.f;
  for (int x = 0; x < 128; ++x) {
    int ph = (kk * x) & 127;
    float th = (2.0f * PI_F / 128.0f) * (float)ph;
    sr += __cosf(th) * row[x];
    si -= __sinf(th) * row[x];
  }
  float2* zin = (float2*)(ws + WS_ZINIT);
  zin[(b * 128 + k) * 32 + h] = make_float2(sr, si);
}

// ---------------------------------------------------------------------------
// Forward: per (b,i,t-chunk of 64): build Hu tile in LDS (as B-fragments),
// then H_ft = F @ Hu via WMMA (2 real GEMMs, real input).
__global__ void __launch_bounds__(256) k_fwd(const float* __restrict__ xi,
                                             float* ws) {
  extern __shared__ float lds[];
  float* Ff   = lds;           // 32768 : F fragments (Re+Im)
  float* HuF  = lds + 32768;   // 8192  : Hu B-fragments
  float* Gl   = lds + 40960;   // 1024
  float* Fu0l = lds + 41984;   // 128
  int tid = threadIdx.x, bid = blockIdx.x;
  int tc = bid & 3, bi = bid >> 2;
  int i = bi & 31, b = bi >> 5;

#ifdef USE_TDM
  if (tid == 0) tdm_copy_f32(ws + WS_FFWD, Ff, 32768u, 1u);
#else
  for (int s = tid; s < 32768; s += 256) Ff[s] = ws[WS_FFWD + s];
#endif
  for (int s = tid; s < 1024; s += 256) Gl[s] = ws[WS_G + bi * 1024 + s];
  if (tid < 128) Fu0l[tid] = ws[WS_FU0 + bi * 128 + tid];
  __syncthreads();

  // Hu[x,t] = Fu0[x] + sum_f G[f,x]*xi[b,f,x,t]; store in B-fragment order.
  for (int e = tid; e < 8192; e += 256) {
    int x = e >> 6, tl = e & 63, t = tc * 64 + tl;
    float hu = Fu0l[x];
    for (int f = 0; f < 8; ++f)
      hu += Gl[f * 128 + x] * xi[((b * 8 + f) * 128 + x) * 256 + t];
    int kt = x >> 2, p = x & 1, h2 = (x >> 1) & 1;
    int lane_ = h2 * 16 + (tl & 15);
    int nt = tl >> 4;
    HuF[((kt * 4 + nt) * 32 + lane_) * 2 + p] = hu;
  }
#ifdef USE_TDM
  TDM_WAIT();
#endif
  __syncthreads();

  int w = tid >> 5, lane = tid & 31;
  const v2f* Ff2 = (const v2f*)Ff;
  const v2f* Hu2 = (const v2f*)HuF;
  float2* hft = (float2*)(ws + WS_HFT);
  for (int nt = 0; nt < 4; ++nt) {
    v8f aR = {0.f, 0.f, 0.f, 0.f, 0.f, 0.f, 0.f, 0.f};
    v8f aI = {0.f, 0.f, 0.f, 0.f, 0.f, 0.f, 0.f, 0.f};
    for (int kt = 0; kt < 32; ++kt) {
      v2f ar = Ff2[((0 * 8 + w) * 32 + kt) * 32 + lane];
      v2f ai = Ff2[((1 * 8 + w) * 32 + kt) * 32 + lane];
      v2f bb = Hu2[(kt * 4 + nt) * 32 + lane];
      aR = wmma_f32(ar, bb, aR);
      aI = wmma_f32(ai, bb, aI);
    }
    int t = tc * 64 + nt * 16 + (lane & 15);
    for (int r = 0; r < 8; ++r) {
      int kmode = 16 * w + r + 8 * (lane >> 4);
      hft[((b * 128 + kmode) * 256 + t) * 32 + i] = make_float2(aR[r], aI[r]);
    }
  }
}

// ---------------------------------------------------------------------------
// Scan: one block per x. W[x] (Re+Im) in LDS as A-fragments (TDM DMA).
// Per step: dz = W (.) (dx (x) z), batched: 2 GEMMs (32x1024)x(1024x16),
// columns n = [y_r(b=0..7) | y_i(b=0..7)].  z <- z + dz, written in place.
__global__ void __launch_bounds__(128) k_scan(const float* __restrict__ Wr,
                                              const float* __restrict__ Wi,
                                              float* ws) {
  extern __shared__ float lds[];
  float* Wfrag = lds;            // 65536 floats = 256KB
  float* dxr = lds + 65536;      // [c][b] -> c*8+b
  float* dxi = lds + 65792;
  float* zr  = lds + 66048;      // [j][b] -> j*8+b
  float* zi  = lds + 66304;
  float* Cst = lds + 66560;      // 4 waves x 16x16

  int tid = threadIdx.x;
  int x = blockIdx.x;

#ifdef USE_TDM
  if (tid == 0) tdm_copy_f32(ws + WS_WFRAG + x * 65536, Wfrag, 32768u, 2u);
#else
  // Fallback: swizzle W into A-fragment layout directly from global.
  for (int s = tid; s < 65536; s += 128) {
    int p = s & 1;
    int lane_ = (s >> 1) & 31;
    int kt = (s >> 6) & 255;
    int m = (s >> 14) & 1;
    int g = s >> 15;
    int i = m * 16 + (lane_ & 15);
    int k = kt * 4 + 2 * (lane_ >> 4) + p;
    const float* Wg = g ? Wi : Wr;
    Wfrag[s] = Wg[x * 32768 + i * 1024 + k];
  }
#endif

  float2* hft = (float2*)(ws + WS_HFT);
  const float2* zin = (const float2*)(ws + WS_ZINIT);
  float hpr[2], hpi[2];
  for (int q = 0; q < 2; ++q) {
    int e = tid * 2 + q;
    int b = e >> 5, i = e & 31;
    float2 h0 = hft[((b * 128 + x) * 256 + 0) * 32 + i];
    hpr[q] = h0.x; hpi[q] = h0.y;
    float2 zv = zin[(b * 128 + x) * 32 + i];
    zr[i * 8 + b] = zv.x; zi[i * 8 + b] = zv.y;
    hft[((b * 128 + x) * 256 + 0) * 32 + i] = zv;   // z_all[t=0] = z_init
  }
#ifdef USE_TDM
  TDM_WAIT();
#endif
  __syncthreads();

  int w = tid >> 5, lane = tid & 31;
  int g = w >> 1, m = w & 1;       // wave role: g in {P=W_r, Q=W_i}, m = row tile
  int bcol = lane & 7;
  int part = (lane >> 3) & 1;      // col n<8 -> y_r, n>=8 -> y_i
  const v2f* Wf2 = (const v2f*)Wfrag;

  for (int t = 0; t < 255; ++t) {
    float hnx[2], hny[2];
    for (int q = 0; q < 2; ++q) {
      int e = tid * 2 + q;
      int b = e >> 5, i = e & 31;
      float2 hn = hft[((b * 128 + x) * 256 + (t + 1)) * 32 + i];
      if (t + 2 < 256)   // hide latency of the next control-path row
        __builtin_prefetch((const void*)&hft[((b * 128 + x) * 256 + (t + 2)) * 32 + i], 0, 1);
      dxr[i * 8 + b] = hn.x - hpr[q];
      dxi[i * 8 + b] = hn.y - hpi[q];
      hnx[q] = hn.x; hny[q] = hn.y;
    }
    __syncthreads();

    v8f acc = {0.f, 0.f, 0.f, 0.f, 0.f, 0.f, 0.f, 0.f};
    for (int kt = 0; kt < 256; ++kt) {
      v2f a = Wf2[((g * 2 + m) * 256 + kt) * 32 + lane];
      int kk = kt * 4 + 2 * (lane >> 4);
      int c = kk >> 5, j = kk & 31;          // c constant over the K-tile
      float dr = dxr[c * 8 + bcol], di = dxi[c * 8 + bcol];
      float z0r = zr[j * 8 + bcol], z0i = zi[j * 8 + bcol];
      float z1r = zr[(j + 1) * 8 + bcol], z1i = zi[(j + 1) * 8 + bcol];
      v2f bb;
      if (part == 0) { bb[0] = dr * z0r - di * z0i; bb[1] = dr * z1r - di * z1i; }
      else           { bb[0] = dr * z0i + di * z0r; bb[1] = dr * z1i + di * z1r; }
      acc = wmma_f32(a, bb, acc);
    }
    for (int r = 0; r < 8; ++r) {
      int row = r + 8 * (lane >> 4);
      Cst[w * 256 + row * 16 + (lane & 15)] = acc[r];
    }
    __syncthreads();

    // dz_r = P[i,b] - Q[i,b+8];  dz_i = P[i,b+8] + Q[i,b]
    for (int q = 0; q < 2; ++q) {
      int e = tid * 2 + q;
      int b = e >> 5, i = e & 31;
      int ihi = i >> 4, ilo = i & 15;
      float Pb  = Cst[ihi * 256 + ilo * 16 + b];
      float Pb8 = Cst[ihi * 256 + ilo * 16 + b + 8];
      float Qb  = Cst[(2 + ihi) * 256 + ilo * 16 + b];
      float Qb8 = Cst[(2 + ihi) * 256 + ilo * 16 + b + 8];
      float znr = zr[i * 8 + b] + (Pb - Qb8);
      float zni = zi[i * 8 + b] + (Pb8 + Qb);
      zr[i * 8 + b] = znr; zi[i * 8 + b] = zni;
      hft[((b * 128 + x) * 256 + (t + 1)) * 32 + i] = make_float2(znr, zni);
      hpr[q] = hnx[q]; hpi[q] = hny[q];
    }
    // next iteration's pre-GEMM barrier orders z/Cst reuse
  }
}

// ---------------------------------------------------------------------------
// Inverse: u[b,h,x,t] = Finv_r @ Z_r + Finv_i @ (-Z_i)  (real part only)
__global__ void __launch_bounds__(256) k_inv(float* ws, float* __restrict__ out) {
  extern __shared__ float lds[];
  float* Fi  = lds;            // 32768
  float* ZrF = lds + 32768;    // 8192
  float* ZiF = lds + 40960;    // 8192 (negated imag)
  int tid = threadIdx.x, bid = blockIdx.x;
  int tc = bid & 3, bh = bid >> 2;
  int h = bh & 31, b = bh >> 5;

#ifdef USE_TDM
  if (tid == 0) tdm_copy_f32(ws + WS_FINV, Fi, 32768u, 1u);
#else
  for (int s = tid; s < 32768; s += 256) Fi[s] = ws[WS_FINV + s];
#endif
  const float2* hft = (const float2*)(ws + WS_HFT);
  for (int e = tid; e < 8192; e += 256) {
    int kidx = e >> 6, tl = e & 63;
    float2 v = hft[((b * 128 + kidx) * 256 + tc * 64 + tl) * 32 + h];
    int kt = kidx >> 2, p = kidx & 1, h2 = (kidx >> 1) & 1;
    int lane_ = h2 * 16 + (tl & 15);
    int nt = tl >> 4;
    int pos = ((kt * 4 + nt) * 32 + lane_) * 2 + p;
    ZrF[pos] = v.x;
    ZiF[pos] = -v.y;
  }
#ifdef USE_TDM
  TDM_WAIT();
#endif
  __syncthreads();

  int w = tid >> 5, lane = tid & 31;
  const v2f* Fi2 = (const v2f*)Fi;
  const v2f* Zr2 = (const v2f*)ZrF;
  const v2f* Zi2 = (const v2f*)ZiF;
  for (int nt = 0; nt < 4; ++nt) {
    v8f acc = {0.f, 0.f, 0.f, 0.f, 0.f, 0.f, 0.f, 0.f};
    for (int kt = 0; kt < 32; ++kt) {
      v2f ar = Fi2[((0 * 8 + w) * 32 + kt) * 32 + lane];
      v2f ai = Fi2[((1 * 8 + w) * 32 + kt) * 32 + lane];
      v2f br = Zr2[(kt * 4 + nt) * 32 + lane];
      v2f bi = Zi2[(kt * 4 + nt) * 32 + lane];
      acc = wmma_f32(ar, br, acc);
      acc = wmma_f32(ai, bi, acc);
    }
    int t = tc * 64 + nt * 16 + (lane & 15);
    for (int r = 0; r < 8; ++r) {
      int xo = 16 * w + r + 8 * (lane >> 4);
      out[((b * 32 + h) * 128 + xo) * 256 + t] = acc[r];
    }
  }
}

// ---------------------------------------------------------------------------
extern "C" void kernel_launch(void* const* d_in, const int* in_sizes, int n_in,
                              void* d_out, int out_size, void* d_ws,
                              size_t ws_size, hipStream_t stream) {
  const float* z0 = (const float*)d_in[0];
  const float* xi = (const float*)d_in[1];
  const float* A  = (const float*)d_in[2];
  const float* Gw = (const float*)d_in[3];
  const float* Wr = (const float*)d_in[4];
  const float* Wi = (const float*)d_in[5];
  float* ws  = (float*)d_ws;
  float* out = (float*)d_out;

  const int LDS_FWD  = (32768 + 8192 + 1024 + 128) * 4;  // 168448
  const int LDS_SCAN = (65536 + 4 * 256 + 1024) * 4;     // 270336
  const int LDS_INV  = (32768 + 8192 + 8192) * 4;        // 196608
  hipFuncSetAttribute((const void*)k_fwd,
                      hipFuncAttributeMaxDynamicSharedMemorySize, LDS_FWD);
  hipFuncSetAttribute((const void*)k_scan,
                      hipFuncAttributeMaxDynamicSharedMemorySize, LDS_SCAN);
  hipFuncSetAttribute((const void*)k_inv,
                      hipFuncAttributeMaxDynamicSharedMemorySize, LDS_INV);

  k_init_dft<<<256, 256, 0, stream>>>(ws);
  k_pre<<<1152, 256, 0, stream>>>(A, Gw, z0, ws);
  k_wfrag<<<32768, 256, 0, stream>>>(Wr, Wi, ws);
  k_z0ft<<<256, 128, 0, stream>>>(z0, ws);
  k_fwd<<<1024, 256, LDS_FWD, stream>>>(xi, ws);
  k_scan<<<128, 128, LDS_SCAN, stream>>>(Wr, Wi, ws);
  k_inv<<<1024, 256, LDS_INV, stream>>>(ws, out);
}